// CBAM_26568667693551
// MI455X (gfx1250) — compile-verified
//
#include <hip/hip_runtime.h>
#include <math.h>

// ---------------------------------------------------------------------------
// CBAM pillar kernel for MI455X (gfx1250, wave32).
// Core GEMMs run on V_WMMA_F32_16X16X4_F32 (full f32 precision).
// One pillar per wave; 4 waves per block to amortize LDS weight staging.
// ---------------------------------------------------------------------------

typedef __attribute__((ext_vector_type(2))) float v2f;
typedef __attribute__((ext_vector_type(8))) float v8f;

#define BINS 32
#define CIN 5
#define KPAD 8          // layer-1 K padded to 8 (two K=4 steps)
#define CHID 32
#define COUT 64
#define CRED 16         // channel-attention hidden (64/4)
#define KSP 7           // bin-attention conv kernel
#define WAVES 4         // pillars (waves) per block
#define SRC_STRIDE 9    // 8 + pad  -> conflict-free
#define HID_STRIDE 33   // 32 + pad
#define X_STRIDE 65     // 64 + pad

__global__ void slot_scatter_kernel(const int* __restrict__ coords,
                                    const int* __restrict__ unq_inv,
                                    int* __restrict__ slots, int n) {
    int i = blockIdx.x * blockDim.x + threadIdx.x;
    if (i < n) {
        int b = coords[i * 4 + 1];
        slots[unq_inv[i] * BINS + b] = i;
    }
}

__device__ __forceinline__ float sigm(float x) {
    return 1.0f / (1.0f + expf(-x));
}

__global__ __launch_bounds__(WAVES * 32) void cbam_main_kernel(
    const float* __restrict__ vf, const int* __restrict__ slots,
    const int* __restrict__ unq_cnt,
    const float* __restrict__ W1, const float* __restrict__ b1,
    const float* __restrict__ W2, const float* __restrict__ b2,
    const float* __restrict__ Wc1, const float* __restrict__ bc1,
    const float* __restrict__ Wc2, const float* __restrict__ bc2,
    const float* __restrict__ Wsp, const float* __restrict__ bsp,
    float* __restrict__ out, int U) {

    // ---- shared weights (block-wide) ----
    __shared__ float s_W1[KPAD * CHID];
    __shared__ float s_b1[CHID];
    __shared__ float s_W2[CHID * COUT];
    __shared__ float s_b2[COUT];
    __shared__ float s_Wc1[COUT * CRED];
    __shared__ float s_bc1[CRED];
    __shared__ float s_Wc2[CRED * COUT];
    __shared__ float s_bc2[COUT];
    __shared__ float s_Wsp[2 * KSP];
    __shared__ float s_bsp;
    // ---- per-wave tiles ----
    __shared__ float s_src[WAVES][BINS * SRC_STRIDE];
    __shared__ float s_hid[WAVES][BINS * HID_STRIDE];
    __shared__ float s_x[WAVES][BINS * X_STRIDE];
    __shared__ float s_pa[WAVES][COUT];
    __shared__ float s_pm[WAVES][COUT];
    __shared__ float s_h[WAVES][2 * CRED];
    __shared__ float s_sm[WAVES][BINS];
    __shared__ float s_sx[WAVES][BINS];
    __shared__ float s_att[WAVES][BINS];

    const int tid = threadIdx.x;
    const int nthr = WAVES * 32;

    for (int i = tid; i < KPAD * CHID; i += nthr) {
        int r = i / CHID, c = i % CHID;
        s_W1[i] = (r < CIN) ? W1[r * CHID + c] : 0.0f;
    }
    for (int i = tid; i < CHID; i += nthr) s_b1[i] = b1[i];
    for (int i = tid; i < CHID * COUT; i += nthr) s_W2[i] = W2[i];
    for (int i = tid; i < COUT; i += nthr) s_b2[i] = b2[i];
    for (int i = tid; i < COUT * CRED; i += nthr) s_Wc1[i] = Wc1[i];
    for (int i = tid; i < CRED; i += nthr) s_bc1[i] = bc1[i];
    for (int i = tid; i < CRED * COUT; i += nthr) s_Wc2[i] = Wc2[i];
    for (int i = tid; i < COUT; i += nthr) s_bc2[i] = bc2[i];
    for (int i = tid; i < 2 * KSP; i += nthr) s_Wsp[i] = Wsp[i];
    if (tid == 0) s_bsp = bsp[0];
    __syncthreads();

    const int w = tid >> 5;        // wave (pillar slot) in block
    const int lane = tid & 31;
    const int half = lane >> 4;    // lane half for WMMA fragment layout
    const int l16 = lane & 15;
    const int p = blockIdx.x * WAVES + w;
    const bool active = (p < U);   // wave-uniform -> EXEC stays all-1 for WMMA

    float* sp = s_src[w];
    float* hd = s_hid[w];
    float* xb = s_x[w];
    const int c0 = lane, c1 = lane + 32;   // channels handled by this lane

    // ---- stage 1: gather voxel features into the 32-bin tile ----
    if (active) {
        int v = slots[p * BINS + lane];    // lane == bin
        float f0 = 0, f1 = 0, f2 = 0, f3 = 0, f4 = 0;
        if (v >= 0) {
            const float* s = vf + (long)v * CIN;
            f0 = s[0]; f1 = s[1]; f2 = s[2]; f3 = s[3]; f4 = s[4];
        }
        float* row = sp + lane * SRC_STRIDE;
        row[0] = f0; row[1] = f1; row[2] = f2; row[3] = f3; row[4] = f4;
        row[5] = 0.0f; row[6] = 0.0f; row[7] = 0.0f; row[8] = 0.0f;
    }
    __syncthreads();

    // ---- stage 2: layer-1 GEMM  (32x8) @ (8x32)  via WMMA f32 16x16x4 ----
    if (active) {
        for (int mt = 0; mt < 2; ++mt) {
            // Hoist A fragments (reused across both N-tiles).
            // A 16x4: lanes 0-15 hold K=k0,k0+1 ; lanes 16-31 hold K=k0+2,k0+3
            v2f afr[2];
#pragma unroll
            for (int kk = 0; kk < 2; ++kk) {
                int k0 = kk * 4 + half * 2;
                afr[kk].x = sp[(mt * 16 + l16) * SRC_STRIDE + k0];
                afr[kk].y = sp[(mt * 16 + l16) * SRC_STRIDE + k0 + 1];
            }
            for (int ct = 0; ct < 2; ++ct) {
                v8f acc = {0, 0, 0, 0, 0, 0, 0, 0};
#pragma unroll
                for (int kk = 0; kk < 2; ++kk) {
                    int k0 = kk * 4 + half * 2;
                    // B 4x16: same K split across lane halves, N = ct*16+l16
                    v2f b;
                    b.x = s_W1[k0 * CHID + ct * 16 + l16];
                    b.y = s_W1[(k0 + 1) * CHID + ct * 16 + l16];
                    acc = __builtin_amdgcn_wmma_f32_16x16x4_f32(
                        false, afr[kk], false, b, (short)0, acc, false, false);
                }
                float bias = s_b1[ct * 16 + l16];
#pragma unroll
                for (int j = 0; j < 8; ++j) {
                    float vv = acc[j] + bias;              // C/D: row = j (+8 hi half)
                    vv = vv > 0.0f ? vv : 0.0f;            // relu
                    hd[(mt * 16 + half * 8 + j) * HID_STRIDE + ct * 16 + l16] = vv;
                }
            }
        }
    }
    __syncthreads();

    // ---- stage 3: layer-2 GEMM  (32x32) @ (32x64) via WMMA ----
    if (active) {
        for (int mt = 0; mt < 2; ++mt) {
            // Hoist all 8 A fragments for this M-tile (reused across 4 N-tiles).
            v2f afr[8];
#pragma unroll
            for (int kk = 0; kk < 8; ++kk) {
                int k0 = kk * 4 + half * 2;
                afr[kk].x = hd[(mt * 16 + l16) * HID_STRIDE + k0];
                afr[kk].y = hd[(mt * 16 + l16) * HID_STRIDE + k0 + 1];
            }
            for (int ct = 0; ct < 4; ++ct) {
                v8f acc = {0, 0, 0, 0, 0, 0, 0, 0};
#pragma unroll
                for (int kk = 0; kk < 8; ++kk) {
                    int k0 = kk * 4 + half * 2;
                    v2f b;
                    b.x = s_W2[k0 * COUT + ct * 16 + l16];
                    b.y = s_W2[(k0 + 1) * COUT + ct * 16 + l16];
                    acc = __builtin_amdgcn_wmma_f32_16x16x4_f32(
                        false, afr[kk], false, b, (short)0, acc, false, false);
                }
                float bias = s_b2[ct * 16 + l16];
#pragma unroll
                for (int j = 0; j < 8; ++j) {
                    xb[(mt * 16 + half * 8 + j) * X_STRIDE + ct * 16 + l16] =
                        acc[j] + bias;
                }
            }
        }
    }
    __syncthreads();

    // ---- stage 4: channel pooling (avg & max over bins) ----
    if (active) {
        float s0 = 0, s1 = 0, m0 = -INFINITY, m1 = -INFINITY;
        for (int b = 0; b < BINS; ++b) {
            float x0 = xb[b * X_STRIDE + c0];
            float x1 = xb[b * X_STRIDE + c1];
            s0 += x0; s1 += x1;
            m0 = fmaxf(m0, x0); m1 = fmaxf(m1, x1);
        }
        s_pa[w][c0] = s0 * (1.0f / BINS);
        s_pa[w][c1] = s1 * (1.0f / BINS);
        s_pm[w][c0] = m0;
        s_pm[w][c1] = m1;
    }
    __syncthreads();

    // ---- stage 5: channel-attention hidden layer (64->16, relu) ----
    if (active) {
        const float* pv = half ? s_pm[w] : s_pa[w];  // low half: avg, high: max
        float hs = s_bc1[l16];
        for (int c = 0; c < COUT; ++c) hs += pv[c] * s_Wc1[c * CRED + l16];
        s_h[w][half * CRED + l16] = hs > 0.0f ? hs : 0.0f;
    }
    __syncthreads();

    // ---- stage 6: channel attention out (16->64), sigmoid, scale x ----
    if (active) {
        float a0 = 2.0f * s_bc2[c0];   // bc2 added in both mlp(avg) and mlp(max)
        float a1 = 2.0f * s_bc2[c1];
        for (int j = 0; j < CRED; ++j) {
            float hj = s_h[w][j] + s_h[w][CRED + j];
            a0 += hj * s_Wc2[j * COUT + c0];
            a1 += hj * s_Wc2[j * COUT + c1];
        }
        float att0 = sigm(a0), att1 = sigm(a1);
        for (int b = 0; b < BINS; ++b) {
            xb[b * X_STRIDE + c0] *= att0;
            xb[b * X_STRIDE + c1] *= att1;
        }
    }
    __syncthreads();

    // ---- stage 7: bin pooling (mean & max over channels) ----
    if (active) {
        float ss = 0, mx = -INFINITY;
        for (int c = 0; c < COUT; ++c) {
            float xv = xb[lane * X_STRIDE + c];
            ss += xv; mx = fmaxf(mx, xv);
        }
        s_sm[w][lane] = ss * (1.0f / COUT);
        s_sx[w][lane] = mx;
    }
    __syncthreads();

    // ---- stage 8: bin attention conv1d (2->1, k=7, pad 3) + sigmoid ----
    if (active) {
        float at = s_bsp;
#pragma unroll
        for (int t = 0; t < KSP; ++t) {
            int idx = lane + t - KSP / 2;
            if (idx >= 0 && idx < BINS)
                at += s_Wsp[t] * s_sm[w][idx] + s_Wsp[KSP + t] * s_sx[w][idx];
        }
        s_att[w][lane] = sigm(at);
    }
    __syncthreads();

    // ---- stage 9: final max over bins; write features + mask ----
    if (active) {
        float M0 = -INFINITY, M1 = -INFINITY;
        for (int b = 0; b < BINS; ++b) {
            float sg = s_att[w][b];
            M0 = fmaxf(M0, xb[b * X_STRIDE + c0] * sg);
            M1 = fmaxf(M1, xb[b * X_STRIDE + c1] * sg);
        }
        out[(long)p * COUT + c0] = M0;
        out[(long)p * COUT + c1] = M1;
        if (lane == 0)
            out[(long)U * COUT + p] = (unq_cnt[p] >= 2) ? 1.0f : 0.0f;
    }
}

extern "C" void kernel_launch(void* const* d_in, const int* in_sizes, int n_in,
                              void* d_out, int out_size, void* d_ws, size_t ws_size,
                              hipStream_t stream) {
    const float* vf      = (const float*)d_in[0];
    const int*   coords  = (const int*)d_in[1];
    // d_in[2] = unq_coords (unused; in_sizes[2] gives U)
    const int*   unq_inv = (const int*)d_in[3];
    const int*   unq_cnt = (const int*)d_in[4];
    const float* W1  = (const float*)d_in[5];
    const float* b1  = (const float*)d_in[6];
    const float* W2  = (const float*)d_in[7];
    const float* b2  = (const float*)d_in[8];
    const float* Wc1 = (const float*)d_in[9];
    const float* bc1 = (const float*)d_in[10];
    const float* Wc2 = (const float*)d_in[11];
    const float* bc2 = (const float*)d_in[12];
    const float* Wsp = (const float*)d_in[13];
    const float* bsp = (const float*)d_in[14];

    const int U = in_sizes[2];
    const int N = in_sizes[3];

    int* slots = (int*)d_ws;                    // U*BINS ints
    size_t slot_bytes = (size_t)U * BINS * sizeof(int);
    // 0xFF bytes == -1 per int; memsetAsync is graph-capture safe.
    hipMemsetAsync(slots, 0xFF, slot_bytes, stream);

    slot_scatter_kernel<<<(N + 255) / 256, 256, 0, stream>>>(coords, unq_inv,
                                                             slots, N);

    int blocks = (U + WAVES - 1) / WAVES;
    cbam_main_kernel<<<blocks, WAVES * 32, 0, stream>>>(
        vf, slots, unq_cnt, W1, b1, W2, b2, Wc1, bc1, Wc2, bc2, Wsp, bsp,
        (float*)d_out, U);
}